// HeteroGCNLinkPredictor_66391604462410
// MI455X (gfx1250) — compile-verified
//
#include <hip/hip_runtime.h>
#include <math.h>

#define H 128
#define EPSBN 1e-5f

typedef float v2f __attribute__((ext_vector_type(2)));
typedef float v8f __attribute__((ext_vector_type(8)));

// LDS weight tile: 32 k-pairs x 128 cols (padded) or 64 k-pairs x 64 cols
#define SW_ELEMS 4160   // max(32*130, 64*65) float2 -> 33.3 KB
#define PITCH128 130    // row pitch (float2) for 128-col tiles
#define PITCH64  65     // row pitch (float2) for 64-col tiles

// ---------------------------------------------------------------- utilities
__global__ __launch_bounds__(256) void k_zero(float* __restrict__ p, long n) {
  long i = (long)blockIdx.x * blockDim.x + threadIdx.x;
  long st = (long)gridDim.x * blockDim.x;
  for (; i < n; i += st) p[i] = 0.0f;
}

// ------------------------------------------------- scatter-sum (mean aggr.)
// one wave (32 lanes) per edge; each lane handles 4 consecutive channels.
__global__ __launch_bounds__(256) void k_scatter(
    const float* __restrict__ xsrc, const int* __restrict__ src,
    const int* __restrict__ dst, float* __restrict__ agg,
    float* __restrict__ cnt, int E) {
  int tid  = blockIdx.x * blockDim.x + threadIdx.x;
  int e    = tid >> 5;
  int lane = tid & 31;
  if (e >= E) return;
  int s = src[e], d = dst[e];
  const float4 v = ((const float4*)(xsrc + (size_t)s * H))[lane];
  float* ag = agg + (size_t)d * H + lane * 4;
  atomicAdd(ag + 0, v.x);
  atomicAdd(ag + 1, v.y);
  atomicAdd(ag + 2, v.z);
  atomicAdd(ag + 3, v.w);
  if (lane == 0) atomicAdd(&cnt[d], 1.0f);
}

// ---------------------------------------------------- SAGE GEMM (in place)
// out[i,:] = (agg[i,:]*1/max(cnt,1)) @ Wl^T + bl + xdst[i,:] @ Wr^T
// One wave per 32-row block (2 independent 16-row accumulator chains sharing
// each LDS B fragment). Weights staged in LDS (transposed, k-pair major) in
// 32KB chunks. All agg reads precede writes -> in-place safe.
__global__ __launch_bounds__(256) void k_sage_gemm(
    float* __restrict__ agg, const float* __restrict__ cnt,
    const float* __restrict__ xdst, const float* __restrict__ Wl,
    const float* __restrict__ bl, const float* __restrict__ Wr, int nrows) {
  __shared__ float2 sW[SW_ELEMS];
  const int tid  = threadIdx.x;
  const int wave = (blockIdx.x * blockDim.x + tid) >> 5;
  const int lane = tid & 31;
  const int i0   = wave * 32;
  const bool active = (i0 < nrows);            // wave-uniform
  const int r    = lane & 15;
  const int ksel = lane >> 4;                  // K pair select in group-of-4
  const int row0 = active ? min(i0 + r, nrows - 1) : 0;
  const int row1 = active ? min(i0 + 16 + r, nrows - 1) : 0;
  const float inv0 = 1.0f / fmaxf(cnt[row0], 1.0f);
  const float inv1 = 1.0f / fmaxf(cnt[row1], 1.0f);
  const float* aRow0 = agg  + (size_t)row0 * H;
  const float* aRow1 = agg  + (size_t)row1 * H;
  const float* xRow0 = xdst + (size_t)row0 * H;
  const float* xRow1 = xdst + (size_t)row1 * H;

  v8f acc0[8], acc1[8];
#pragma unroll
  for (int t = 0; t < 8; ++t) {
    float bv = bl[t * 16 + r];
    acc0[t] = (v8f){bv, bv, bv, bv, bv, bv, bv, bv};
    acc1[t] = acc0[t];
  }

  for (int term = 0; term < 2; ++term) {
    const float* W    = (term == 0) ? Wl : Wr;
    const float* src0 = (term == 0) ? aRow0 : xRow0;
    const float* src1 = (term == 0) ? aRow1 : xRow1;
    const float  s0   = (term == 0) ? inv0 : 1.0f;
    const float  s1   = (term == 0) ? inv1 : 1.0f;
    for (int half = 0; half < 2; ++half) {
      __syncthreads();                         // previous tile fully consumed
      // stage W[col][half*64 + 2p .. +1] -> sW[p][col] (coalesced global read)
      for (int idx = tid; idx < 4096; idx += 256) {
        int col = idx >> 5, p = idx & 31;
        const float* g = W + (size_t)col * H + half * 64 + 2 * p;
        sW[p * PITCH128 + col] = make_float2(g[0], g[1]);
      }
      __syncthreads();
      if (active) {
        for (int kk = 0; kk < 64; kk += 4) {
          const int kg = half * 64 + kk + ksel * 2;
          v2f a0, a1;
          a0.x = src0[kg] * s0;
          a0.y = src0[kg + 1] * s0;
          a1.x = src1[kg] * s1;
          a1.y = src1[kg + 1] * s1;
          const int prow = (kk >> 1) + ksel;
#pragma unroll
          for (int t = 0; t < 8; ++t) {
            float2 bw = sW[prow * PITCH128 + t * 16 + r];
            v2f b; b.x = bw.x; b.y = bw.y;
            acc0[t] = __builtin_amdgcn_wmma_f32_16x16x4_f32(
                false, a0, false, b, (short)0, acc0[t], false, false);
            acc1[t] = __builtin_amdgcn_wmma_f32_16x16x4_f32(
                false, a1, false, b, (short)0, acc1[t], false, false);
          }
        }
      }
    }
  }
  // write back over agg (C/D layout: VGPR v -> M=v (+8 for upper lanes))
  if (active) {
    const int mofs = ksel * 8;
#pragma unroll
    for (int t = 0; t < 8; ++t)
#pragma unroll
      for (int v = 0; v < 8; ++v) {
        int oa = i0 + v + mofs;
        int ob = i0 + 16 + v + mofs;
        if (oa < nrows) agg[(size_t)oa * H + t * 16 + r] = acc0[t][v];
        if (ob < nrows) agg[(size_t)ob * H + t * 16 + r] = acc1[t][v];
      }
  }
}

// ------------------------------------------------------------ BatchNorm1d
__global__ void k_bn_stats(const float* __restrict__ x, int rows, int C,
                           float* __restrict__ stats) {
  int c = threadIdx.x;
  float s = 0.0f, s2 = 0.0f;
  for (int r = blockIdx.x; r < rows; r += gridDim.x) {
    float v = x[(size_t)r * C + c];
    s += v;
    s2 += v * v;
  }
  atomicAdd(&stats[c], s);
  atomicAdd(&stats[C + c], s2);
}

__global__ void k_bn_finalize(float* __restrict__ stats, int C, float inv_n,
                              const float* __restrict__ gamma,
                              const float* __restrict__ beta) {
  int c = threadIdx.x;
  float m   = stats[c] * inv_n;
  float var = stats[C + c] * inv_n - m * m;        // biased variance
  float sc  = gamma[c] * rsqrtf(var + EPSBN);
  stats[c]     = sc;                               // scale
  stats[C + c] = beta[c] - m * sc;                 // shift
}

__global__ __launch_bounds__(256) void k_bn_apply_relu(
    float* __restrict__ x, long n, int cmask, int C,
    const float* __restrict__ stats) {
  long i  = (long)blockIdx.x * blockDim.x + threadIdx.x;
  long st = (long)gridDim.x * blockDim.x;
  for (; i < n; i += st) {
    int c = (int)(i & cmask);
    x[i] = fmaxf(x[i] * stats[c] + stats[C + c], 0.0f);
  }
}

// ------------------------------------------ decoder GEMM1 (fused gather)
// h1[l,o] = sum_k concat(u[idx0[l]], j[idx1[l]])[k] * W1[o][k] + b1[o]
// W1 is [128][256]; staged in 4 LDS chunks of 64 K values. M=32 per wave.
__global__ __launch_bounds__(256) void k_dec_gemm1(
    const float* __restrict__ u, const float* __restrict__ j,
    const int* __restrict__ eli, const float* __restrict__ W1,
    const float* __restrict__ b1, float* __restrict__ h1, int L) {
  __shared__ float2 sW[SW_ELEMS];
  const int tid  = threadIdx.x;
  const int wave = (blockIdx.x * blockDim.x + tid) >> 5;
  const int lane = tid & 31;
  const int i0   = wave * 32;
  const bool active = (i0 < L);
  const int r    = lane & 15;
  const int ksel = lane >> 4;
  const int row0 = active ? min(i0 + r, L - 1) : 0;
  const int row1 = active ? min(i0 + 16 + r, L - 1) : 0;
  const float* zu0 = u + (size_t)eli[row0] * H;
  const float* zj0 = j + (size_t)eli[L + row0] * H;
  const float* zu1 = u + (size_t)eli[row1] * H;
  const float* zj1 = j + (size_t)eli[L + row1] * H;

  v8f acc0[8], acc1[8];
#pragma unroll
  for (int t = 0; t < 8; ++t) {
    float bv = b1[t * 16 + r];
    acc0[t] = (v8f){bv, bv, bv, bv, bv, bv, bv, bv};
    acc1[t] = acc0[t];
  }
  for (int q = 0; q < 4; ++q) {                // K chunk [q*64, q*64+64)
    __syncthreads();
    for (int idx = tid; idx < 4096; idx += 256) {
      int col = idx >> 5, p = idx & 31;
      const float* g = W1 + (size_t)col * (2 * H) + q * 64 + 2 * p;
      sW[p * PITCH128 + col] = make_float2(g[0], g[1]);
    }
    __syncthreads();
    if (active) {
      const float* src0 = (q < 2) ? zu0 : zj0;
      const float* src1 = (q < 2) ? zu1 : zj1;
      const int base = (q & 1) * 64;
      for (int kk = 0; kk < 64; kk += 4) {
        const int kg = base + kk + ksel * 2;
        v2f a0, a1;
        a0.x = src0[kg];
        a0.y = src0[kg + 1];
        a1.x = src1[kg];
        a1.y = src1[kg + 1];
        const int prow = (kk >> 1) + ksel;
#pragma unroll
        for (int t = 0; t < 8; ++t) {
          float2 bw = sW[prow * PITCH128 + t * 16 + r];
          v2f b; b.x = bw.x; b.y = bw.y;
          acc0[t] = __builtin_amdgcn_wmma_f32_16x16x4_f32(
              false, a0, false, b, (short)0, acc0[t], false, false);
          acc1[t] = __builtin_amdgcn_wmma_f32_16x16x4_f32(
              false, a1, false, b, (short)0, acc1[t], false, false);
        }
      }
    }
  }
  if (active) {
    const int mofs = ksel * 8;
#pragma unroll
    for (int t = 0; t < 8; ++t)
#pragma unroll
      for (int v = 0; v < 8; ++v) {
        int oa = i0 + v + mofs;
        int ob = i0 + 16 + v + mofs;
        if (oa < L) h1[(size_t)oa * H + t * 16 + r] = acc0[t][v];
        if (ob < L) h1[(size_t)ob * H + t * 16 + r] = acc1[t][v];
      }
  }
}

// ------------------------------------------ decoder GEMM2: [L,128]x[128,64]
__global__ __launch_bounds__(256) void k_dec_gemm2(
    const float* __restrict__ h1, const float* __restrict__ W2,
    const float* __restrict__ b2, float* __restrict__ h2, int L) {
  __shared__ float2 sW[SW_ELEMS];              // 64 k-pairs x 64 cols used
  const int tid  = threadIdx.x;
  const int wave = (blockIdx.x * blockDim.x + tid) >> 5;
  const int lane = tid & 31;
  const int i0   = wave * 32;
  const bool active = (i0 < L);
  const int r    = lane & 15;
  const int ksel = lane >> 4;
  const int row0 = active ? min(i0 + r, L - 1) : 0;
  const int row1 = active ? min(i0 + 16 + r, L - 1) : 0;
  const float* aRow0 = h1 + (size_t)row0 * H;
  const float* aRow1 = h1 + (size_t)row1 * H;

  // stage W2[col][2p..2p+1] -> sW[p][col]  (W2 is [64][128] row-major)
  for (int idx = tid; idx < 4096; idx += 256) {
    int col = idx >> 6, p = idx & 63;
    const float* g = W2 + (size_t)col * H + 2 * p;
    sW[p * PITCH64 + col] = make_float2(g[0], g[1]);
  }
  __syncthreads();

  v8f acc0[4], acc1[4];
#pragma unroll
  for (int t = 0; t < 4; ++t) {
    float bv = b2[t * 16 + r];
    acc0[t] = (v8f){bv, bv, bv, bv, bv, bv, bv, bv};
    acc1[t] = acc0[t];
  }
  if (active) {
    for (int kk = 0; kk < H; kk += 4) {
      const int kg = kk + ksel * 2;
      v2f a0, a1;
      a0.x = aRow0[kg];
      a0.y = aRow0[kg + 1];
      a1.x = aRow1[kg];
      a1.y = aRow1[kg + 1];
      const int prow = (kk >> 1) + ksel;
#pragma unroll
      for (int t = 0; t < 4; ++t) {
        float2 bw = sW[prow * PITCH64 + t * 16 + r];
        v2f b; b.x = bw.x; b.y = bw.y;
        acc0[t] = __builtin_amdgcn_wmma_f32_16x16x4_f32(
            false, a0, false, b, (short)0, acc0[t], false, false);
        acc1[t] = __builtin_amdgcn_wmma_f32_16x16x4_f32(
            false, a1, false, b, (short)0, acc1[t], false, false);
      }
    }
    const int mofs = ksel * 8;
#pragma unroll
    for (int t = 0; t < 4; ++t)
#pragma unroll
      for (int v = 0; v < 8; ++v) {
        int oa = i0 + v + mofs;
        int ob = i0 + 16 + v + mofs;
        if (oa < L) h2[(size_t)oa * 64 + t * 16 + r] = acc0[t][v];
        if (ob < L) h2[(size_t)ob * 64 + t * 16 + r] = acc1[t][v];
      }
  }
}

// ------------------------------------------ final matvec + sigmoid
__global__ __launch_bounds__(256) void k_dec_out(
    const float* __restrict__ h2, const float* __restrict__ W3,
    const float* __restrict__ b3, float* __restrict__ out, int L) {
  int l = blockIdx.x * blockDim.x + threadIdx.x;
  if (l >= L) return;
  const float4* hr = (const float4*)(h2 + (size_t)l * 64);
  const float4* w4 = (const float4*)W3;
  float s = b3[0];
#pragma unroll
  for (int k = 0; k < 16; ++k) {
    float4 h = hr[k], w = w4[k];
    s += h.x * w.x + h.y * w.y + h.z * w.z + h.w * w.w;
  }
  out[l] = 1.0f / (1.0f + __expf(-s));
}

// =========================================================== host sequence
extern "C" void kernel_launch(void* const* d_in, const int* in_sizes, int n_in,
                              void* d_out, int out_size, void* d_ws,
                              size_t ws_size, hipStream_t stream) {
  const float* x_user = (const float*)d_in[0];
  const float* x_job  = (const float*)d_in[1];
  const int* e_u2j    = (const int*)d_in[2];   // [0]=src(user), [1]=dst(job)
  const int* e_j2u    = (const int*)d_in[3];   // [0]=src(job),  [1]=dst(user)
  const int* eli      = (const int*)d_in[4];   // [0]=user idx,  [1]=job idx
  const float* Wl_u2j = (const float*)d_in[5];
  const float* bl_u2j = (const float*)d_in[6];
  const float* Wr_u2j = (const float*)d_in[7];
  const float* Wl_j2u = (const float*)d_in[8];
  const float* bl_j2u = (const float*)d_in[9];
  const float* Wr_j2u = (const float*)d_in[10];
  const float* bn_g   = (const float*)d_in[11];
  const float* bn_b   = (const float*)d_in[12];
  const float* W1 = (const float*)d_in[13];
  const float* b1 = (const float*)d_in[14];
  const float* g1 = (const float*)d_in[15];
  const float* be1 = (const float*)d_in[16];
  const float* W2 = (const float*)d_in[17];
  const float* b2 = (const float*)d_in[18];
  const float* g2 = (const float*)d_in[19];
  const float* be2 = (const float*)d_in[20];
  const float* W3 = (const float*)d_in[21];
  const float* b3 = (const float*)d_in[22];

  const int Nn = in_sizes[0] / H;    // 150000
  const int E  = in_sizes[2] / 2;    // 600000
  const int L  = in_sizes[4] / 2;    // 200000

  // workspace layout (floats): Pu | Pj | Qu | Qj | cntA | cntB | stats
  // decode reuses Pu/Pj region for h1 (L*H) and h2 (L*64): exactly 2*N*H.
  float* ws = (float*)d_ws;
  const size_t NHf = (size_t)Nn * H;
  float* Pu = ws;
  float* Pj = ws + NHf;
  float* Qu = ws + 2 * NHf;
  float* Qj = ws + 3 * NHf;
  float* cntA = ws + 4 * NHf;
  float* cntB = cntA + Nn;
  float* stats = cntB + Nn;          // 2*H floats, reused per BN
  float* h1 = ws;                    // L*H
  float* h2 = ws + (size_t)L * H;    // L*64

  const int scatBlk = (E * 32 + 255) / 256;
  const int sageBlk = ((Nn + 31) / 32 + 7) / 8;
  const int decBlk  = ((L + 31) / 32 + 7) / 8;

  auto run_bn = [&](float* x, int rows, int C, const float* g,
                    const float* b) {
    k_zero<<<1, 256, 0, stream>>>(stats, 2 * C);
    k_bn_stats<<<512, C, 0, stream>>>(x, rows, C, stats);
    k_bn_finalize<<<1, C, 0, stream>>>(stats, C, 1.0f / (float)rows, g, b);
    long tot = (long)rows * C;
    k_bn_apply_relu<<<2048, 256, 0, stream>>>(x, tot, C - 1, C, stats);
  };

  const float* cu = x_user;
  const float* cj = x_job;
  float* nu[2] = {Pu, Qu};
  float* nj[2] = {Pj, Qj};

  for (int layer = 0; layer < 2; ++layer) {
    float* au = nu[layer];   // aggregation target for users (j2u)
    float* aj = nj[layer];   // aggregation target for jobs  (u2j)
    k_zero<<<2048, 256, 0, stream>>>(au, (long)NHf);
    k_zero<<<2048, 256, 0, stream>>>(aj, (long)NHf);
    k_zero<<<256, 256, 0, stream>>>(cntA, Nn);
    k_zero<<<256, 256, 0, stream>>>(cntB, Nn);
    k_scatter<<<scatBlk, 256, 0, stream>>>(cu, e_u2j, e_u2j + E, aj, cntA, E);
    k_scatter<<<scatBlk, 256, 0, stream>>>(cj, e_j2u, e_j2u + E, au, cntB, E);
    const size_t wo = (size_t)layer * H * H;
    const size_t bo = (size_t)layer * H;
    k_sage_gemm<<<sageBlk, 256, 0, stream>>>(aj, cntA, cj, Wl_u2j + wo,
                                             bl_u2j + bo, Wr_u2j + wo, Nn);
    k_sage_gemm<<<sageBlk, 256, 0, stream>>>(au, cntB, cu, Wl_j2u + wo,
                                             bl_j2u + bo, Wr_j2u + wo, Nn);
    run_bn(au, Nn, H, bn_g + bo, bn_b + bo);
    run_bn(aj, Nn, H, bn_g + bo, bn_b + bo);
    cu = au;
    cj = aj;
  }

  // decoder (final embeddings live in Qu/Qj; h1/h2 reuse dead Pu/Pj region)
  k_dec_gemm1<<<decBlk, 256, 0, stream>>>(Qu, Qj, eli, W1, b1, h1, L);
  run_bn(h1, L, H, g1, be1);
  k_dec_gemm2<<<decBlk, 256, 0, stream>>>(h1, W2, b2, h2, L);
  run_bn(h2, L, 64, g2, be2);
  k_dec_out<<<(L + 255) / 256, 256, 0, stream>>>(h2, W3, b3, (float*)d_out, L);
}